// TRNetParallel_78013785964660
// MI455X (gfx1250) — compile-verified
//
#include <hip/hip_runtime.h>
#include <math.h>

// Problem constants (from reference)
constexpr int B = 512, D = 16, N = 64, M = 32, R = 32, C = 10;
constexpr int BT = 16;            // batch elements per workgroup (= waves per block)
constexpr int NW = BT;            // 16 waves * 32 lanes = 512 threads
constexpr int LDS_FLOATS = 2 * BT * R * R;   // core + prod, 32768 floats = 128 KB

typedef __attribute__((ext_vector_type(2))) float v2f;
typedef __attribute__((ext_vector_type(8))) float v8f;

// ---------------------------------------------------------------------------
// Kernel 1: feat[b,d,m] = sum_n tensor[b,d,n] * W[n,m] + bias[m]   (34 MFLOP)
// ---------------------------------------------------------------------------
__global__ void feat_kernel(const float* __restrict__ tensor,
                            const float* __restrict__ W,
                            const float* __restrict__ bias,
                            float* __restrict__ feat) {
    int idx = blockIdx.x * blockDim.x + threadIdx.x;   // over B*D*M
    if (idx >= B * D * M) return;
    int m  = idx % M;
    int bd = idx / M;                                  // b*D + d
    const float* t = tensor + (size_t)bd * N;
    float acc = bias[m];
#pragma unroll 8
    for (int n = 0; n < N; ++n)
        acc = fmaf(t[n], W[n * M + m], acc);
    feat[idx] = acc;
}

// ---------------------------------------------------------------------------
// Kernel 2 (fused main): per (c, b-tile) build 32x32 cores via WMMA and chain
// the per-b matrix product across d in LDS; emit trace -> logits[b,c].
// ---------------------------------------------------------------------------
__global__ void __launch_bounds__(NW * 32)
chain_kernel(const float* __restrict__ feat,
             const float* __restrict__ G,
             float* __restrict__ logits) {
    extern __shared__ float smem[];
    float* lds_core = smem;                    // [BT][R][R]
    float* lds_prod = smem + BT * R * R;       // [BT][R][R]

    const int bt   = blockIdx.x % (B / BT);
    const int c    = blockIdx.x / (B / BT);
    const int lane = threadIdx.x & 31;
    const int wave = threadIdx.x >> 5;
    const int half = lane >> 4;                // 0: lanes 0-15, 1: lanes 16-31
    const int l16  = lane & 15;

    // Init prod = identity (so the d-loop handles d=0 uniformly).
    for (int idx = threadIdx.x; idx < BT * R * R; idx += NW * 32) {
        int ij = idx & (R * R - 1);
        lds_prod[idx] = ((ij >> 5) == (ij & 31)) ? 1.0f : 0.0f;
    }
    __syncthreads();

    for (int d = 0; d < D; ++d) {
        // ---- Phase A: core[b][i][j] = sum_m feat[b,d,m] * G[c,d,i,m,j] ----
        // A-fragment (16x4 f32): M = local b, K = m. Same for all i/j tiles.
        v2f afrag[8];
        {
            const float* fr = feat + ((size_t)(bt * BT + l16) * D + d) * M;
#pragma unroll
            for (int kk = 0; kk < 8; ++kk) {
                int m0 = kk * 4 + 2 * half;
                afrag[kk].x = fr[m0];
                afrag[kk].y = fr[m0 + 1];
            }
        }
#pragma unroll
        for (int ib = 0; ib < 2; ++ib) {
            int i = wave * 2 + ib;             // 16 waves x 2 = all 32 rows i
            const float* Gi = G + ((((size_t)c * D + d) * R + i) * M) * R;
#pragma unroll
            for (int tj = 0; tj < 2; ++tj) {
                int j = tj * 16 + l16;
                v8f acc = {0, 0, 0, 0, 0, 0, 0, 0};
#pragma unroll
                for (int kk = 0; kk < 8; ++kk) {
                    int m0 = kk * 4 + 2 * half;
                    v2f bfrag;                 // B (4x16): K = m, N = j
                    bfrag.x = Gi[(size_t)m0 * R + j];
                    bfrag.y = Gi[(size_t)(m0 + 1) * R + j];
                    acc = __builtin_amdgcn_wmma_f32_16x16x4_f32(
                        false, afrag[kk], false, bfrag, (short)0, acc,
                        false, false);
                }
                // D tile: row r holds (b = r + 8*half, j = tj*16 + l16)
#pragma unroll
                for (int r = 0; r < 8; ++r)
                    lds_core[(r + 8 * half) * R * R + i * R + j] = acc[r];
            }
        }
        __syncthreads();

        // ---- Phase B: per-b chain step  P <- P @ core  (wave w owns b=w) ----
        float* P  = lds_prod + wave * R * R;
        float* Cg = lds_core + wave * R * R;
        v8f pacc[2][2];
        const v8f vzero = {0, 0, 0, 0, 0, 0, 0, 0};
        pacc[0][0] = vzero; pacc[0][1] = vzero;
        pacc[1][0] = vzero; pacc[1][1] = vzero;
#pragma unroll
        for (int k0 = 0; k0 < R; k0 += 4) {
            int kc = k0 + 2 * half;
            v2f a0, a1, b0, b1;
            a0.x = P[l16 * R + kc];          a0.y = P[l16 * R + kc + 1];
            a1.x = P[(16 + l16) * R + kc];   a1.y = P[(16 + l16) * R + kc + 1];
            b0.x = Cg[kc * R + l16];         b0.y = Cg[(kc + 1) * R + l16];
            b1.x = Cg[kc * R + 16 + l16];    b1.y = Cg[(kc + 1) * R + 16 + l16];
            pacc[0][0] = __builtin_amdgcn_wmma_f32_16x16x4_f32(
                false, a0, false, b0, (short)0, pacc[0][0], false, false);
            pacc[0][1] = __builtin_amdgcn_wmma_f32_16x16x4_f32(
                false, a0, false, b1, (short)0, pacc[0][1], false, false);
            pacc[1][0] = __builtin_amdgcn_wmma_f32_16x16x4_f32(
                false, a1, false, b0, (short)0, pacc[1][0], false, false);
            pacc[1][1] = __builtin_amdgcn_wmma_f32_16x16x4_f32(
                false, a1, false, b1, (short)0, pacc[1][1], false, false);
        }
        // Write P back (same-wave LDS ops are in-order, so reads above are safe).
#pragma unroll
        for (int ti = 0; ti < 2; ++ti)
#pragma unroll
            for (int tj = 0; tj < 2; ++tj)
#pragma unroll
                for (int r = 0; r < 8; ++r)
                    P[(ti * 16 + r + 8 * half) * R + tj * 16 + l16] =
                        pacc[ti][tj][r];
        __syncthreads();   // cores are re-written by other waves next d
    }

    // Trace of P for b = wave: lane L holds diag element L.
    float tr = lds_prod[wave * R * R + lane * R + lane];
#pragma unroll
    for (int off = 16; off > 0; off >>= 1)
        tr += __shfl_xor(tr, off, 32);
    if (lane == 0)
        logits[(bt * BT + wave) * C + c] = tr;
}

// ---------------------------------------------------------------------------
// Kernel 3: log_softmax over C=10 per batch row.
// ---------------------------------------------------------------------------
__global__ void softmax_kernel(const float* __restrict__ logits,
                               float* __restrict__ out) {
    int b = blockIdx.x * blockDim.x + threadIdx.x;
    if (b >= B) return;
    float v[C];
    float mx = -INFINITY;
#pragma unroll
    for (int c = 0; c < C; ++c) { v[c] = logits[b * C + c]; mx = fmaxf(mx, v[c]); }
    float s = 0.0f;
#pragma unroll
    for (int c = 0; c < C; ++c) s += __expf(v[c] - mx);
    float lse = mx + __logf(s);
#pragma unroll
    for (int c = 0; c < C; ++c) out[b * C + c] = v[c] - lse;
}

// ---------------------------------------------------------------------------
extern "C" void kernel_launch(void* const* d_in, const int* in_sizes, int n_in,
                              void* d_out, int out_size, void* d_ws, size_t ws_size,
                              hipStream_t stream) {
    const float* tensor = (const float*)d_in[0];   // (B, D, N)
    const float* W_fm   = (const float*)d_in[1];   // (N, M)
    const float* b_fm   = (const float*)d_in[2];   // (M,)
    const float* G      = (const float*)d_in[3];   // (C, D, R, M, R)
    float* out = (float*)d_out;                    // (B, C) f32

    float* feat   = (float*)d_ws;                  // B*D*M floats (1 MB)
    float* logits = feat + (size_t)B * D * M;      // B*C floats

    feat_kernel<<<(B * D * M + 255) / 256, 256, 0, stream>>>(tensor, W_fm, b_fm, feat);

    chain_kernel<<<C * (B / BT), NW * 32, LDS_FLOATS * sizeof(float), stream>>>(
        feat, G, logits);

    softmax_kernel<<<(B + 127) / 128, 128, 0, stream>>>(logits, out);
}